// GCNLinkPred_12421045420213
// MI455X (gfx1250) — compile-verified
//
#include <hip/hip_runtime.h>

typedef float v2f __attribute__((ext_vector_type(2)));
typedef float v8f __attribute__((ext_vector_type(8)));

#define N_NODES 100000
#define KDIM 128

// ---------------- degree / normalization ----------------

__global__ void zero_kernel(float* __restrict__ p, int n) {
  int i = blockIdx.x * blockDim.x + threadIdx.x;
  if (i < n) p[i] = 0.0f;
}

__global__ void deg_kernel(const int* __restrict__ dst, float* __restrict__ deg, int nE) {
  int i = blockIdx.x * blockDim.x + threadIdx.x;
  if (i < nE) atomicAdd(&deg[dst[i]], 1.0f);
}

__global__ void dis_kernel(float* __restrict__ degdis, int n) {
  int i = blockIdx.x * blockDim.x + threadIdx.x;
  if (i < n) degdis[i] = rsqrtf(1.0f + degdis[i]);
}

// ---------------- fp32 WMMA GEMM: C[M,NCOLS] = A[M,128] * W[128,NCOLS] ----------------
// One wave per 16x16 output tile. gridDim.x = M/16 tiles, wave w in block = n-tile w.
// A-frag (16x4 f32): lane m = lane&15, half-wave hi selects K pair {k+2hi, k+2hi+1}.
// B-frag (4x16 f32): lane n = lane&15, half-wave hi selects K pair rows of W.
// C/D (16x16 f32): VGPR r -> M = 8*hi + r, N = lane&15.
// NCOLS is a template parameter so all row strides fold into immediate offsets.

template <int NCOLS>
__global__ void gemm_wmma_kernel(const float* __restrict__ A, const float* __restrict__ W,
                                 float* __restrict__ C) {
  const int lane = threadIdx.x & 31;
  const int wave = threadIdx.x >> 5;
  const int hi   = lane >> 4;
  const int lo   = lane & 15;
  const int m0   = blockIdx.x * 16;
  const int n    = wave * 16 + lo;

  // per-lane base pointers; all subsequent accesses use compile-time offsets
  const float* __restrict__ arow = A + (size_t)(m0 + lo) * KDIM + 2 * hi;
  const float* __restrict__ wcol = W + (size_t)(2 * hi) * NCOLS + n;

  v8f acc = {};
#pragma unroll
  for (int k = 0; k < KDIM; k += 4) {
    v2f a, b;
    a.x = arow[k + 0];
    a.y = arow[k + 1];
    b.x = wcol[(size_t)k * NCOLS];
    b.y = wcol[(size_t)k * NCOLS + NCOLS];
    acc = __builtin_amdgcn_wmma_f32_16x16x4_f32(
        /*neg_a=*/false, a, /*neg_b=*/false, b,
        /*c_mod=*/(short)0, acc, /*reuse_a=*/false, /*reuse_b=*/false);
  }

  float* __restrict__ crow = C + (size_t)(m0 + hi * 8) * NCOLS + n;
#pragma unroll
  for (int r = 0; r < 8; ++r)
    crow[(size_t)r * NCOLS] = acc[r];
}

// ---------------- self-loop + bias init: agg = h * dis[i]^2 + bias[f] ----------------

__global__ void init_agg_kernel(const float* __restrict__ h, const float* __restrict__ dis,
                                const float* __restrict__ bias, float* __restrict__ agg,
                                int total, int F) {
  int i = blockIdx.x * blockDim.x + threadIdx.x;
  if (i >= total) return;
  int node = i / F;
  int f = i - node * F;
  float ds = dis[node];
  agg[i] = h[i] * ds * ds + bias[f];
}

// ---------------- edge gather-scale-scatter ----------------
// F=128: one wave per edge, each lane handles a float4 chunk.
// F=64 : two edges per wave (16 lanes each).

template <int F>
__global__ void scatter_kernel(const float* __restrict__ h, const int* __restrict__ src,
                               const int* __restrict__ dst, const float* __restrict__ dis,
                               float* __restrict__ agg, int nE) {
  const int gw   = blockIdx.x * (blockDim.x >> 5) + (threadIdx.x >> 5);
  const int lane = threadIdx.x & 31;
  constexpr int chunks = F / 4;       // float4 chunks per edge row
  constexpr int epw    = 32 / chunks; // edges per wave
  const int sub = lane / chunks;
  const int l   = lane % chunks;
  const int e = gw * epw + sub;
  if (e >= nE) return;

  const int s = src[e];
  const int d = dst[e];
  const float coef = dis[s] * dis[d];

  const float4 v = *(const float4*)(h + (size_t)s * F + l * 4);
  float* out = agg + (size_t)d * F + l * 4;
  atomicAdd(out + 0, v.x * coef);
  atomicAdd(out + 1, v.y * coef);
  atomicAdd(out + 2, v.z * coef);
  atomicAdd(out + 3, v.w * coef);
}

__global__ void relu_kernel(float* __restrict__ p, int n) {
  int i = blockIdx.x * blockDim.x + threadIdx.x;
  if (i < n) p[i] = fmaxf(p[i], 0.0f);
}

// ---------------- label edge dot products ----------------

__global__ void dot_kernel(const float* __restrict__ z, const int* __restrict__ ls,
                           const int* __restrict__ ld, float* __restrict__ out, int nL) {
  int e = blockIdx.x * blockDim.x + threadIdx.x;
  if (e >= nL) return;
  const float4* a = (const float4*)(z + (size_t)ls[e] * 64);
  const float4* b = (const float4*)(z + (size_t)ld[e] * 64);
  float s = 0.0f;
#pragma unroll
  for (int i = 0; i < 16; ++i) {
    float4 x = a[i], y = b[i];
    s += x.x * y.x + x.y * y.y + x.z * y.z + x.w * y.w;
  }
  out[e] = s;
}

// ---------------- driver ----------------

extern "C" void kernel_launch(void* const* d_in, const int* in_sizes, int n_in,
                              void* d_out, int out_size, void* d_ws, size_t ws_size,
                              hipStream_t stream) {
  const float* x  = (const float*)d_in[0];
  const int*   ei = (const int*)d_in[1];   // edge_index [2, E]
  const int*   el = (const int*)d_in[2];   // edge_label_index [2, L]
  const float* W1 = (const float*)d_in[3];
  const float* b1 = (const float*)d_in[4];
  const float* W2 = (const float*)d_in[5];
  const float* b2 = (const float*)d_in[6];

  const int nE = in_sizes[1] / 2;
  const int nL = in_sizes[2] / 2;
  const int* src = ei;
  const int* dst = ei + nE;
  const int* ls  = el;
  const int* ld  = el + nL;

  // workspace carve (floats): dis[100k] | bufA[N*128] | bufB[N*128]  (~103 MB)
  float* ws   = (float*)d_ws;
  float* dis  = ws;
  float* bufA = ws + 100352;                    // h1, then h2
  float* bufB = bufA + (size_t)N_NODES * 128;   // agg1/hrelu, then z
  float* out  = (float*)d_out;

  const int elem1 = N_NODES * 128;
  const int elem2 = N_NODES * 64;

  // degree -> dis = (1+deg)^-1/2
  zero_kernel<<<(N_NODES + 255) / 256, 256, 0, stream>>>(dis, N_NODES);
  deg_kernel<<<(nE + 255) / 256, 256, 0, stream>>>(dst, dis, nE);
  dis_kernel<<<(N_NODES + 255) / 256, 256, 0, stream>>>(dis, N_NODES);

  // ----- layer 1: h1 = x @ W1 ; agg1 = scatter + self-loop + b1 ; relu -----
  gemm_wmma_kernel<128><<<N_NODES / 16, 256, 0, stream>>>(x, W1, bufA);
  init_agg_kernel<<<(elem1 + 255) / 256, 256, 0, stream>>>(bufA, dis, b1, bufB, elem1, 128);
  scatter_kernel<128><<<(nE + 7) / 8, 256, 0, stream>>>(bufA, src, dst, dis, bufB, nE);
  relu_kernel<<<(elem1 + 255) / 256, 256, 0, stream>>>(bufB, elem1);

  // ----- layer 2: h2 = hrelu @ W2 ; z = scatter + self-loop + b2 -----
  gemm_wmma_kernel<64><<<N_NODES / 16, 128, 0, stream>>>(bufB, W2, bufA);
  init_agg_kernel<<<(elem2 + 255) / 256, 256, 0, stream>>>(bufA, dis, b2, bufB, elem2, 64);
  scatter_kernel<64><<<(nE + 15) / 16, 256, 0, stream>>>(bufA, src, dst, dis, bufB, nE);

  // ----- link scores -----
  dot_kernel<<<(nL + 255) / 256, 256, 0, stream>>>(bufB, ls, ld, out, nL);
}